// QModel_45423574123077
// MI455X (gfx1250) — compile-verified
//
#include <hip/hip_runtime.h>
#include <math.h>
#include <stdint.h>

typedef __attribute__((ext_vector_type(2))) float v2f;
typedef __attribute__((ext_vector_type(8))) float v8f;

#define QDIM 512     // 2^9 state length
#define KDIM 256     // amp length (= x row length)
#define NDIM 1024    // re|im * 512 output columns

__device__ __forceinline__ float2 cmul(float2 a, float2 b) {
    return make_float2(a.x * b.x - a.y * b.y, a.x * b.y + a.y * b.x);
}
__device__ __forceinline__ float2 cadd(float2 a, float2 b) {
    return make_float2(a.x + b.x, a.y + b.y);
}
__device__ __forceinline__ float2 conjc(float2 a) { return make_float2(a.x, -a.y); }

__device__ __forceinline__ void wait_asynccnt0() {
#if __has_builtin(__builtin_amdgcn_s_wait_asynccnt)
    __builtin_amdgcn_s_wait_asynccnt(0);
#else
    asm volatile("s_wait_asynccnt 0" ::: "memory");
#endif
}

// Per-lane async copy: 8 bytes global -> LDS, tracked by ASYNCcnt.
__device__ __forceinline__ void async_g2lds_b64(uint32_t lds_byte_off, const void* gaddr) {
    asm volatile("global_load_async_to_lds_b64 %0, %1, off"
                 :: "v"(lds_byte_off), "v"(gaddr)
                 : "memory");
}

// ---------------------------------------------------------------------------
// Kernel 1: materialize W2 from weights.
// Block k simulates the circuit on basis state e_{2k+1}; 512 threads hold the
// 512-complex state in double-buffered LDS. Result column U[:,2k+1] is stored
// pair-interleaved: W2[(k>>1)*2048 + 2*n + (k&1)], n<512 = Re, n>=512 = Im.
// ---------------------------------------------------------------------------
__global__ void __launch_bounds__(512) build_w_kernel(const float* __restrict__ weights,
                                                      float* __restrict__ W2) {
    __shared__ float2 buf[2][QDIM];
    const int i = threadIdx.x;   // state index 0..511
    const int k = blockIdx.x;    // basis column 0..255
    int cur = 0;

    buf[0][i] = (i == 2 * k + 1) ? make_float2(1.f, 0.f) : make_float2(0.f, 0.f);
    __syncthreads();

    const int cw[8] = {0, 2, 4, 6, 1, 3, 5, 7};
    const int tw[8] = {1, 3, 5, 7, 2, 4, 6, 8};

    for (int layer = 0; layer < 2; ++layer) {
        for (int wire = 0; wire < 9; ++wire) {
            const float* w = weights + (layer * 9 + wire) * 3;
            float s0, c0, s1, c1, s2, c2;
            sincosf(0.5f * w[0], &s0, &c0);
            sincosf(0.5f * w[1], &s1, &c1);
            sincosf(0.5f * w[2], &s2, &c2);
            float2 za = make_float2(c0, s0);
            float2 zc = make_float2(c2, s2);
            float2 cb = make_float2(c1, 0.f);
            float2 sb = make_float2(0.f, -s1);
            float2 u00 = cmul(cmul(conjc(zc), cb), conjc(za));
            float2 u01 = cmul(cmul(conjc(zc), sb), za);
            float2 u10 = cmul(cmul(zc, sb), conjc(za));
            float2 u11 = cmul(cmul(zc, cb), za);

            const int shift = 8 - wire;
            const int stride = 1 << shift;
            const int bit = (i >> shift) & 1;
            float2 mine  = buf[cur][i];
            float2 other = buf[cur][i ^ stride];
            float2 nv = (bit == 0) ? cadd(cmul(u00, mine), cmul(u01, other))
                                   : cadd(cmul(u10, other), cmul(u11, mine));
            buf[cur ^ 1][i] = nv;
            __syncthreads();
            cur ^= 1;
        }
        // Compose the 8 CNOT permutations into one index gather:
        // new[i] = old[p0(p1(...p7(i)...))]
        int j = i;
        for (int m = 7; m >= 0; --m) {
            j ^= ((j >> (8 - cw[m])) & 1) << (8 - tw[m]);
        }
        float2 g = buf[cur][j];
        __syncthreads();
        buf[cur ^ 1][i] = g;
        __syncthreads();
        cur ^= 1;
    }

    float2 s = buf[cur][i];
    float* row = W2 + (size_t)(k >> 1) * (2 * NDIM) + (k & 1);
    row[2 * i]          = s.x;   // Re at n = i
    row[2 * (i + QDIM)] = s.y;   // Im at n = i + 512
}

// ---------------------------------------------------------------------------
// Kernel 2: GEMM  S = x(4096x256) * W(256x1024)  via V_WMMA_F32_16X16X4_F32.
// 8 waves per block share double-buffered 16KB B panels staged into LDS with
// GLOBAL_LOAD_ASYNC_TO_LDS_B64 (ASYNCcnt). Fused |.|^2 parity reduction +
// normalization epilogue.
// ---------------------------------------------------------------------------
__global__ void __launch_bounds__(256) gemm_prob_kernel(const float* __restrict__ x,
                                                        const float* __restrict__ W2,
                                                        float* __restrict__ out) {
    // Double-buffered B panel: [buf][pair-row 0..127][col 0..15] of float2.
    __shared__ __align__(16) v2f panelB[2][128 * 16];

    const int t    = threadIdx.x;          // 0..255
    const int lane = t & 31;
    const int wid  = t >> 5;
    const int mt   = blockIdx.x * 8 + wid; // 0..255 M tiles
    const int rowbase = mt * 16;

    const int m    = lane & 15;            // A-matrix row for this lane
    const int koff = (lane >> 4) << 1;     // 0 or 2: K sub-offset per half-wave

    // Preload entire 16x256 A tile as 64 v2f fragments (128 VGPRs).
    v2f areg[64];
    const float* aptr = x + (size_t)(rowbase + m) * KDIM + koff;
#pragma unroll
    for (int kk = 0; kk < 64; ++kk) {
        areg[kk] = *(const v2f*)(aptr + kk * 4);
    }

    const v2f* wptr = (const v2f*)W2;      // [pair][n] -> {W[2p][n], W[2p+1][n]}
    const int bcolbase = lane & 15;
    const int pairoff  = koff >> 1;        // 0 or 1

    // Staging: thread t copies elements e = t + 256*j (j=0..7) of the 2048-
    // element panel: pair = e>>4 (= (t>>4) + 16j), col = e&15 (= t&15).
    const v2f* gstage = wptr + (size_t)(t >> 4) * NDIM + (t & 15);

    // Stage panel nt=0 into buffer 0.
#pragma unroll
    for (int j = 0; j < 8; ++j) {
        uint32_t ldsoff = (uint32_t)(uintptr_t)(&panelB[0][t + 256 * j]);
        async_g2lds_b64(ldsoff, (const void*)(gstage + (size_t)16 * j * NDIM));
    }
    wait_asynccnt0();
    __syncthreads();

    float pPart[8];
#pragma unroll
    for (int r = 0; r < 8; ++r) pPart[r] = 0.f;

    for (int nt = 0; nt < 64; ++nt) {
        const int pb = nt & 1;

        // Kick off async stage of the next panel into the other buffer.
        if (nt + 1 < 64) {
#pragma unroll
            for (int j = 0; j < 8; ++j) {
                uint32_t ldsoff = (uint32_t)(uintptr_t)(&panelB[pb ^ 1][t + 256 * j]);
                async_g2lds_b64(ldsoff,
                    (const void*)(gstage + (size_t)16 * j * NDIM + (nt + 1) * 16));
            }
        }

        const v2f* lb = &panelB[pb][0];
        v8f acc = {};
#pragma unroll
        for (int kk = 0; kk < 64; ++kk) {
            v2f b = lb[(kk * 2 + pairoff) * 16 + bcolbase];   // ds_load_b64
            acc = __builtin_amdgcn_wmma_f32_16x16x4_f32(
                false, areg[kk], false, b, (short)0, acc, false, false);
        }
#pragma unroll
        for (int r = 0; r < 8; ++r) pPart[r] += acc[r] * acc[r];

        wait_asynccnt0();     // next panel fully in LDS (this wave's share)
        __syncthreads();      // all waves done staging + done reading old panel
    }

    // Each lane's column parity is (lane & 1). Reduce the 8 same-parity lanes
    // in each half-wave, then fetch the opposite parity for the normalizer.
#pragma unroll
    for (int r = 0; r < 8; ++r) {
        float v = pPart[r];
        v += __shfl_xor(v, 2, 32);
        v += __shfl_xor(v, 4, 32);
        v += __shfl_xor(v, 8, 32);
        float tot = v + __shfl_xor(v, 1, 32);   // p'0 + p'1 = ||x_row||^2
        pPart[r] = v / tot;
    }

    if ((lane & 14) == 0) {                 // lanes 0,1,16,17 hold final values
        const int rbase  = rowbase + ((lane >> 4) << 3);   // rows 0-7 or 8-15
        const int parity = lane & 1;
#pragma unroll
        for (int r = 0; r < 8; ++r) {
            out[(size_t)(rbase + r) * 2 + parity] = pPart[r];
        }
    }
}

extern "C" void kernel_launch(void* const* d_in, const int* in_sizes, int n_in,
                              void* d_out, int out_size, void* d_ws, size_t ws_size,
                              hipStream_t stream) {
    (void)in_sizes; (void)n_in; (void)out_size; (void)ws_size;
    const float* x       = (const float*)d_in[0];   // 4096*256 f32
    const float* weights = (const float*)d_in[1];   // 2*9*3 f32
    float* out = (float*)d_out;                     // 4096*2 f32
    float* W2  = (float*)d_ws;                      // 256*1024 f32 = 1 MB scratch

    build_w_kernel<<<256, 512, 0, stream>>>(weights, W2);
    gemm_prob_kernel<<<32, 256, 0, stream>>>(x, W2, out);
}